// UpsamplingAttention_29119878267468
// MI455X (gfx1250) — compile-verified
//
#include <hip/hip_runtime.h>
#include <hip/hip_bf16.h>

// Problem constants (fixed by setup_inputs in the reference):
//   N=8, K=256, C=512, M=512, P=2, T_max=2048
#define NB   8
#define KK   256
#define CC   512
#define MM   512
#define TMAX 2048
#define TT   32          // t-rows per workgroup in the main kernel
#define WSROW 260        // padded row stride for scores/W in LDS

typedef float v2f __attribute__((ext_vector_type(2)));
typedef float v8f __attribute__((ext_vector_type(8)));

static __device__ __forceinline__ float swish_fast(float x) {
    return x / (1.0f + __expf(-x));
}

// ---------------------------------------------------------------------------
// Kernel 1: durations, cumsum (S0/E0), mel_length, mel_mask
// grid: NB blocks x 256 threads
// ---------------------------------------------------------------------------
__global__ void k_dur(const float* __restrict__ dur_out,
                      float* __restrict__ S0, float* __restrict__ E0,
                      int* __restrict__ mel_len, int* __restrict__ mel_mask) {
    int n = blockIdx.x;
    int k = threadIdx.x;
    __shared__ float sc[KK];
    __shared__ int   si[KK];
    __shared__ int   ml;

    float d0 = dur_out[n * KK + k];
    float d  = fmaxf(expf(d0) - 1.0f, 0.0f);
    int   rd = (int)rintf(d);           // round-half-even, matches jnp.round
    if (rd < 1) rd = 1;

    sc[k] = d;
    si[k] = rd;
    __syncthreads();

    // Hillis-Steele inclusive scan over 256 elements
    for (int off = 1; off < KK; off <<= 1) {
        float v = (k >= off) ? sc[k - off] : 0.0f;
        __syncthreads();
        sc[k] += v;
        __syncthreads();
    }
    float inc = sc[k];
    S0[n * KK + k] = inc - d;
    E0[n * KK + k] = inc;

    for (int s = 128; s > 0; s >>= 1) {
        if (k < s) si[k] += si[k + s];
        __syncthreads();
    }
    if (k == 0) { ml = si[0]; mel_len[n] = si[0]; }
    __syncthreads();

    int mel = ml;
    for (int t = k; t < TMAX; t += 256)
        mel_mask[n * TMAX + t] = (t < mel) ? 1 : 0;
}

// ---------------------------------------------------------------------------
// Kernel 2: Conv1D(kernel=3, SAME) [N,K,C] x [3,C,8] -> conv_raw [N,K,8]
// ---------------------------------------------------------------------------
__global__ void k_conv(const float* __restrict__ V,
                       const float* __restrict__ cw,
                       const float* __restrict__ cb,
                       float* __restrict__ conv_raw) {
    int g = blockIdx.x * blockDim.x + threadIdx.x;   // < NB*KK*8
    int n   = g >> 11;
    int rem = g & 2047;
    int k   = rem >> 3;
    int ch  = rem & 7;

    float acc = cb[ch];
    for (int w = 0; w < 3; ++w) {
        int kk = k + w - 1;
        if (kk < 0 || kk >= KK) continue;
        const float* vp = V + ((size_t)(n * KK + kk)) * CC;
        const float* wp = cw + (size_t)w * CC * 8 + ch;
#pragma unroll 4
        for (int c = 0; c < CC; ++c)
            acc += vp[c] * wp[c * 8];
    }
    conv_raw[g] = acc;
}

// ---------------------------------------------------------------------------
// Kernel 3: BN stats (deterministic tree reduce) + BN + swish, then fold the
// conv features into the first MLP layers:
//   g1[p,j]  = b1[j]  + sum_i convV[p,i]*w1[2+i,j]   (scoring layer-1 bias)
//   ga[p,q]  = ab1[q] + sum_i convV[p,i]*aw1[2+i,q]  (aux layer-1 bias)
// (w1/aw1 row 10 corresponds to a feature the reference never provides ->
//  treated as exactly 0.)
// single block, 256 threads
// ---------------------------------------------------------------------------
__global__ void k_bnpre(const float* __restrict__ conv_raw,
                        const float* __restrict__ gam,
                        const float* __restrict__ bet,
                        const float* __restrict__ w1,  const float* __restrict__ b1,
                        const float* __restrict__ aw1, const float* __restrict__ ab1,
                        float* __restrict__ g1, float* __restrict__ ga) {
    __shared__ float rs[8][256];
    __shared__ float rq[8][256];
    __shared__ float sca[8], shf[8];
    __shared__ float w1c[128];   // w1 rows 2..9  ([8 conv][16])
    __shared__ float b1_s[16];
    __shared__ float aw1c[16];   // aw1 rows 2..9 ([8 conv][2])
    __shared__ float ab1_s[2];
    int tid = threadIdx.x;

    if (tid < 128) w1c[tid] = w1[32 + tid];
    if (tid < 16)  { b1_s[tid] = b1[tid]; aw1c[tid] = aw1[4 + tid]; }
    if (tid < 2)   ab1_s[tid] = ab1[tid];

    float s[8] = {0}, q[8] = {0};
    for (int p = tid; p < NB * KK; p += 256) {
        const float* row = conv_raw + p * 8;
#pragma unroll
        for (int ch = 0; ch < 8; ++ch) { float v = row[ch]; s[ch] += v; q[ch] += v * v; }
    }
#pragma unroll
    for (int ch = 0; ch < 8; ++ch) { rs[ch][tid] = s[ch]; rq[ch][tid] = q[ch]; }
    __syncthreads();

    for (int st = 128; st > 0; st >>= 1) {
        if (tid < st) {
#pragma unroll
            for (int ch = 0; ch < 8; ++ch) {
                rs[ch][tid] += rs[ch][tid + st];
                rq[ch][tid] += rq[ch][tid + st];
            }
        }
        __syncthreads();
    }
    if (tid < 8) {
        const float invn = 1.0f / (float)(NB * KK);
        float mean = rs[tid][0] * invn;
        float var  = rq[tid][0] * invn - mean * mean;
        float a    = gam[tid] * rsqrtf(var + 1e-3f);
        sca[tid]   = a;
        shf[tid]   = bet[tid] - mean * a;
    }
    __syncthreads();

    for (int p = tid; p < NB * KK; p += 256) {
        const float* row = conv_raw + p * 8;
        float y[8];
#pragma unroll
        for (int ch = 0; ch < 8; ++ch)
            y[ch] = swish_fast(sca[ch] * row[ch] + shf[ch]);

#pragma unroll
        for (int j = 0; j < 16; ++j) {
            float a = b1_s[j];
#pragma unroll
            for (int ch = 0; ch < 8; ++ch) a += y[ch] * w1c[ch * 16 + j];
            g1[p * 16 + j] = a;
        }
#pragma unroll
        for (int p2 = 0; p2 < 2; ++p2) {
            float a = ab1_s[p2];
#pragma unroll
            for (int ch = 0; ch < 8; ++ch) a += y[ch] * aw1c[ch * 2 + p2];
            ga[p * 2 + p2] = a;
        }
    }
}

// ---------------------------------------------------------------------------
// Kernel 4: fused S/E + scoring MLP + softmax + aux/ctx + WMMA GEMM.
// Fully-masked t-blocks (tb >= mel_length): softmax input is identically 0
// -> W = 1/256 uniform exactly; the scoring MLP and softmax are skipped.
// grid: (T/TT=64, N=8) blocks x 256 threads (8 waves)
// ---------------------------------------------------------------------------
__global__ __launch_bounds__(256)
void k_main(const float* __restrict__ V,  const float* __restrict__ pm,
            const float* __restrict__ S0, const float* __restrict__ E0,
            const int*   __restrict__ mel_len,
            const float* __restrict__ g1, const float* __restrict__ ga,
            const float* __restrict__ w1,
            const float* __restrict__ w2, const float* __restrict__ b2,
            const float* __restrict__ w3, const float* __restrict__ b3,
            const float* __restrict__ aw1,
            const float* __restrict__ aw2, const float* __restrict__ ab2,
            const float* __restrict__ projw,
            float* __restrict__ out) {
    __shared__ float w01_s[32];                 // w1 rows 0 (S) and 1 (E)
    __shared__ float w2_s[256], b2_s[16], w3_s[16];
    __shared__ float aw01_s[4], aw2_s[4], ab2_s[2];
    __shared__ float b3_s;
    __shared__ float g1_s[KK * 17];             // stride-17 pad (bank-conflict-free)
    __shared__ float ga_s[KK * 3];              // stride-3 pad
    __shared__ float S0_s[KK], E0_s[KK], pm_s[KK];
    __shared__ __align__(16) float W_s[TT * WSROW];
    __shared__ float ctxpart[256][2];
    __shared__ float ctx_s[TT][2];
    __shared__ int   mel_s;

    int tid = threadIdx.x;
    int n   = blockIdx.y;
    int tb  = blockIdx.x * TT;

    // ---- stage into LDS --------------------------------------------------
    if (tid < 32) w01_s[tid] = w1[tid];
    w2_s[tid] = w2[tid];
    if (tid < 16) { b2_s[tid] = b2[tid]; w3_s[tid] = w3[tid]; }
    if (tid < 4)  { aw01_s[tid] = aw1[tid]; aw2_s[tid] = aw2[tid]; }
    if (tid < 2)  ab2_s[tid] = ab2[tid];
    if (tid == 0) { b3_s = b3[0]; mel_s = mel_len[n]; }
    S0_s[tid] = S0[n * KK + tid];
    E0_s[tid] = E0[n * KK + tid];
    pm_s[tid] = pm[n * KK + tid];
    for (int i = tid; i < KK * 2; i += 256)
        ga_s[(i >> 1) * 3 + (i & 1)] = ga[n * KK * 2 + i];
    __syncthreads();

    int  mel  = mel_s;
    bool fast = (tb >= mel);                    // block-uniform

    if (!fast) {
        for (int i = tid; i < KK * 16; i += 256)
            g1_s[(i >> 4) * 17 + (i & 15)] = g1[n * KK * 16 + i];
    }
    __syncthreads();

    int   t_local = tid >> 3;                   // 0..31
    int   sub     = tid & 7;                    // 0..7
    float melf    = ((tb + t_local) < mel) ? 1.0f : 0.0f;
    float Tg      = (float)(tb + t_local + 1);

    // ---- phase 1: scoring MLP (only where some t-row is live) -----------
    if (!fast) {
        for (int i = 0; i < 32; ++i) {
            int   k   = i * 8 + sub;
            float att = melf * pm_s[k];
            float x0  = (Tg - S0_s[k]) * att;
            float x1  = (E0_s[k] - Tg) * att;

            float h1[16];
#pragma unroll
            for (int j = 0; j < 16; ++j) {
                float a = g1_s[k * 17 + j] + x0 * w01_s[j] + x1 * w01_s[16 + j];
                h1[j] = swish_fast(a);
            }
            float sc = b3_s;
#pragma unroll
            for (int j = 0; j < 16; ++j) {
                float a = b2_s[j];
#pragma unroll
                for (int ii = 0; ii < 16; ++ii) a += h1[ii] * w2_s[ii * 16 + j];
                sc += swish_fast(a) * w3_s[j];
            }
            W_s[t_local * WSROW + k] = sc;
        }
    }
    __syncthreads();

    // ---- phase 2: softmax (general) or exact uniform 1/256 (fast) -------
    if (!fast) {
        if (tid < TT) {
            int   t  = tid;
            float mf = ((tb + t) < mel) ? 1.0f : 0.0f;
            float mx = -3.402823466e38f;
            for (int k = 0; k < KK; ++k) {
                float v = -1000000.0f * (mf * pm_s[k]) * W_s[t * WSROW + k];
                mx = fmaxf(mx, v);
            }
            float ssum = 0.0f;
            for (int k = 0; k < KK; ++k) {
                float v = -1000000.0f * (mf * pm_s[k]) * W_s[t * WSROW + k];
                float e = __expf(v - mx);
                W_s[t * WSROW + k] = e;
                ssum += e;
            }
            float inv = 1.0f / ssum;
            for (int k = 0; k < KK; ++k) W_s[t * WSROW + k] *= inv;
        }
    } else {
        const float u = 1.0f / 256.0f;          // exp(0)/sum(exp(0)) exactly
        for (int r = 0; r < TT; ++r) W_s[r * WSROW + tid] = u;
    }
    __syncthreads();

    // ---- phase 3: aux MLP + ctx ----------------------------------------
    if (!fast) {
        float c0a = 0.0f, c1a = 0.0f;
        for (int i = 0; i < 32; ++i) {
            int   k   = i * 8 + sub;
            float att = melf * pm_s[k];
            float x0  = (Tg - S0_s[k]) * att;
            float x1  = (E0_s[k] - Tg) * att;

            float a0 = swish_fast(ga_s[k * 3 + 0] + x0 * aw01_s[0] + x1 * aw01_s[2]);
            float a1 = swish_fast(ga_s[k * 3 + 1] + x0 * aw01_s[1] + x1 * aw01_s[3]);
            float c0 = swish_fast(ab2_s[0] + a0 * aw2_s[0] + a1 * aw2_s[2]);
            float c1 = swish_fast(ab2_s[1] + a0 * aw2_s[1] + a1 * aw2_s[3]);
            float w  = W_s[t_local * WSROW + k];
            c0a += w * c0;
            c1a += w * c1;
        }
        ctxpart[tid][0] = c0a;                  // tid == t_local*8 + sub
        ctxpart[tid][1] = c1a;
        __syncthreads();
        if (tid < TT * 2) {
            int t = tid >> 1, p = tid & 1;
            float s2 = 0.0f;
#pragma unroll
            for (int j = 0; j < 8; ++j) s2 += ctxpart[t * 8 + j][p];
            ctx_s[t][p] = s2;
        }
    } else {
        // x0=x1=0 exactly; all t rows share the same ctx with weight 1/256
        int k = tid;
        float a0 = swish_fast(ga_s[k * 3 + 0]);
        float a1 = swish_fast(ga_s[k * 3 + 1]);
        ctxpart[tid][0] = swish_fast(ab2_s[0] + a0 * aw2_s[0] + a1 * aw2_s[2]);
        ctxpart[tid][1] = swish_fast(ab2_s[1] + a0 * aw2_s[1] + a1 * aw2_s[3]);
        __syncthreads();
        for (int s2 = 128; s2 > 0; s2 >>= 1) {  // deterministic tree reduce
            if (tid < s2) {
                ctxpart[tid][0] += ctxpart[tid + s2][0];
                ctxpart[tid][1] += ctxpart[tid + s2][1];
            }
            __syncthreads();
        }
        if (tid < TT * 2)
            ctx_s[tid >> 1][tid & 1] = ctxpart[0][tid & 1] * (1.0f / 256.0f);
    }
    __syncthreads();

    // ---- phase 4: O = W @ V_n + ctx @ proj_w via V_WMMA_F32_16X16X4_F32 -
    {
        int wv   = tid >> 5;            // wave id: owns 64 output columns
        int lane = tid & 31;
        int half = lane >> 4;           // K pair {0,1} vs {2,3}
        int lcol = lane & 15;           // tile column / A-row M

        const float* vbase = V + (size_t)n * KK * CC;
        for (int ct = 0; ct < 4; ++ct) {
            int c0 = wv * 64 + ct * 16;
            v8f acc0 = {};
            v8f acc1 = {};
            const float* vptr = vbase + c0 + lcol;
            for (int ks = 0; ks < 64; ++ks) {
                int krow = ks * 4 + half * 2;
                v2f b;
                b.x = vptr[(size_t)krow * CC];
                b.y = vptr[(size_t)(krow + 1) * CC];
                v2f a0 = *(const v2f*)(&W_s[lcol * WSROW + krow]);
                v2f a1 = *(const v2f*)(&W_s[(16 + lcol) * WSROW + krow]);
                acc0 = __builtin_amdgcn_wmma_f32_16x16x4_f32(
                    false, a0, false, b, (short)0, acc0, false, false);
                acc1 = __builtin_amdgcn_wmma_f32_16x16x4_f32(
                    false, a1, false, b, (short)0, acc1, false, false);
            }
            int   c   = c0 + lcol;
            float pw0 = projw[c];
            float pw1 = projw[MM + c];
#pragma unroll
            for (int r = 0; r < 8; ++r) {
                int   m    = r + half * 8;
                float add0 = ctx_s[m][0] * pw0 + ctx_s[m][1] * pw1;
                out[((size_t)(n * TMAX + tb + m)) * MM + c]      = acc0[r] + add0;
                float add1 = ctx_s[16 + m][0] * pw0 + ctx_s[16 + m][1] * pw1;
                out[((size_t)(n * TMAX + tb + 16 + m)) * MM + c] = acc1[r] + add1;
            }
        }
    }
}

// ---------------------------------------------------------------------------
extern "C" void kernel_launch(void* const* d_in, const int* in_sizes, int n_in,
                              void* d_out, int out_size, void* d_ws, size_t ws_size,
                              hipStream_t stream) {
    (void)in_sizes; (void)n_in; (void)out_size; (void)ws_size;

    const float* V      = (const float*)d_in[0];
    const float* durin  = (const float*)d_in[1];
    const float* pm     = (const float*)d_in[2];
    const float* convw  = (const float*)d_in[3];
    const float* convb  = (const float*)d_in[4];
    const float* gam    = (const float*)d_in[5];
    const float* bet    = (const float*)d_in[6];
    const float* w1     = (const float*)d_in[7];
    const float* b1     = (const float*)d_in[8];
    const float* w2     = (const float*)d_in[9];
    const float* b2     = (const float*)d_in[10];
    const float* w3     = (const float*)d_in[11];
    const float* b3     = (const float*)d_in[12];
    const float* aw1    = (const float*)d_in[13];
    const float* ab1    = (const float*)d_in[14];
    const float* aw2    = (const float*)d_in[15];
    const float* ab2    = (const float*)d_in[16];
    const float* projw  = (const float*)d_in[17];
    // d_in[18] = T_max scalar (hardcoded 2048, matches setup_inputs)

    float* out = (float*)d_out;
    float* ws  = (float*)d_ws;

    // workspace layout (floats)
    float* S0       = ws;                         // [2048]
    float* E0       = ws + 2048;                  // [2048]
    int*   mel_len  = (int*)(ws + 4096);          // [8] (+pad)
    float* conv_raw = ws + 4112;                  // [16384]
    float* g1       = conv_raw + NB * KK * 8;     // [NB*KK*16] = 32768
    float* ga       = g1 + NB * KK * 16;          // [NB*KK*2]  = 4096

    int* mel_mask = ((int*)d_out) + (size_t)NB * TMAX * MM;

    k_dur<<<NB, 256, 0, stream>>>(durin, S0, E0, mel_len, mel_mask);
    k_conv<<<(NB * KK * 8) / 256, 256, 0, stream>>>(V, convw, convb, conv_raw);
    k_bnpre<<<1, 256, 0, stream>>>(conv_raw, gam, bet, w1, b1, aw1, ab1, g1, ga);

    dim3 grid(TMAX / TT, NB);
    k_main<<<grid, 256, 0, stream>>>(V, pm, S0, E0, mel_len, g1, ga,
                                     w1, w2, b2, w3, b3,
                                     aw1, aw2, ab2, projw, out);
}